// NeuralGARCHLayer_46926812676813
// MI455X (gfx1250) — compile-verified
//
#include <hip/hip_runtime.h>
#include <stdint.h>
#include <math.h>

// GARCH(1,1) scan over [B=8192, S=4096] f32 + 96-step forecast.
// Memory-bound: 128 MiB stream -> ~5.7 us floor at 23.3 TB/s.
// One wave32 per row; 128-float chunks staged via CDNA5 async-to-LDS
// (ASYNCcnt) with an 8-deep pipeline; ds_swizzle butterfly reduction;
// affine-recurrence chunk combine h = be^128 * h + sum(be^(127-p) * c_p).

#define WAVES_PER_BLOCK 8
#define SEQ   4096
#define CHUNK 128                 // floats per chunk (512 B per wave-load)
#define NCHUNK (SEQ / CHUNK)      // 32
#define DEPTH 8                   // async pipeline depth (LDS buffers/wave)

#define WAIT_ASYNC(n) asm volatile("s_wait_asynccnt " #n ::: "memory")

__device__ __forceinline__ void issue_async_chunk(const float* row, unsigned lds_base,
                                                  int ck, int lane) {
    const float* g = row + ck * CHUNK + lane * 4;                       // 16 B / lane
    unsigned l = lds_base + (unsigned)((ck & (DEPTH - 1)) * CHUNK * 4)  // buffer select
                          + (unsigned)(lane * 16);
    asm volatile("global_load_async_to_lds_b128 %0, %1, off"
                 :: "v"(l), "v"(g) : "memory");
}

__device__ __forceinline__ float wave32_sum(float v) {
    // XOR butterfly via ds_swizzle (group-of-32 mode): all lanes end with the total.
    v += __int_as_float(__builtin_amdgcn_ds_swizzle(__float_as_int(v), 0x041F)); // xor 1
    v += __int_as_float(__builtin_amdgcn_ds_swizzle(__float_as_int(v), 0x081F)); // xor 2
    v += __int_as_float(__builtin_amdgcn_ds_swizzle(__float_as_int(v), 0x101F)); // xor 4
    v += __int_as_float(__builtin_amdgcn_ds_swizzle(__float_as_int(v), 0x201F)); // xor 8
    v += __int_as_float(__builtin_amdgcn_ds_swizzle(__float_as_int(v), 0x401F)); // xor 16
    return v;
}

__global__ __launch_bounds__(WAVES_PER_BLOCK * 32)
void garch_scan_forecast(const float* __restrict__ returns,
                         const float* __restrict__ omega,
                         const float* __restrict__ alpha,
                         const float* __restrict__ beta,
                         const float* __restrict__ h0p,
                         const int*   __restrict__ pred_len,
                         float*       __restrict__ out,
                         int B)
{
    __shared__ float smem[WAVES_PER_BLOCK * DEPTH * CHUNK]; // 32 KB / block
    const int lane = threadIdx.x & 31;
    const int wave = threadIdx.x >> 5;
    const int b = blockIdx.x * WAVES_PER_BLOCK + wave;
    if (b >= B) return;

    // Constrained coefficients (match reference softplus/sigmoid)
    const float om = log1pf(expf(omega[0]));
    const float al = 0.2f / (1.0f + expf(-alpha[0]));
    const float be = 0.8f / (1.0f + expf(-beta[0]));
    float       h  = log1pf(expf(h0p[0]));
    const int   P  = pred_len[0];

    const float be2 = be * be, be3 = be2 * be;
    const float wl  = powf(be, (float)(CHUNK - 4 - 4 * lane)); // be^(124-4*lane)
    const float beC = powf(be, (float)CHUNK);                  // be^128

    const float* row = returns + (size_t)b * SEQ;
    float* mybuf = smem + wave * (DEPTH * CHUNK);
    const unsigned lds_base = (unsigned)(uintptr_t)mybuf; // low 32 bits = LDS offset

    // Prologue: fill the async pipeline
    #pragma unroll
    for (int i = 0; i < DEPTH; ++i)
        issue_async_chunk(row, lds_base, i, lane);

    for (int ck = 0; ck < NCHUNK; ++ck) {
        // Wait until chunk ck has landed (loads complete in order):
        // allow (#chunks issued after ck, capped at DEPTH-1) to stay outstanding.
        int rem = NCHUNK - 1 - ck;
        if (rem > DEPTH - 1) rem = DEPTH - 1;
        switch (rem) {
            case 7: WAIT_ASYNC(0x7); break;
            case 6: WAIT_ASYNC(0x6); break;
            case 5: WAIT_ASYNC(0x5); break;
            case 4: WAIT_ASYNC(0x4); break;
            case 3: WAIT_ASYNC(0x3); break;
            case 2: WAIT_ASYNC(0x2); break;
            case 1: WAIT_ASYNC(0x1); break;
            default: WAIT_ASYNC(0x0); break;
        }

        const float4 r = *reinterpret_cast<const float4*>(
            mybuf + (ck & (DEPTH - 1)) * CHUNK + lane * 4);
        // Ensure the LDS read has returned to VGPRs before the DMA reuses this buffer.
        asm volatile("s_wait_dscnt 0x0" ::: "memory");

        if (ck + DEPTH < NCHUNK)
            issue_async_chunk(row, lds_base, ck + DEPTH, lane);

        // c_t = om + al * r_t^2 ; lane partial = be^(124-4l) * (c0*be^3 + c1*be^2 + c2*be + c3)
        float c0 = fmaf(al * r.x, r.x, om);
        float c1 = fmaf(al * r.y, r.y, om);
        float c2 = fmaf(al * r.z, r.z, om);
        float c3 = fmaf(al * r.w, r.w, om);
        float part = wl * fmaf(c0, be3, fmaf(c1, be2, fmaf(c2, be, c3)));
        part = wave32_sum(part);
        h = fmaf(beC, h, part);   // h <- be^128 * h + weighted chunk sum
    }

    // Forecast: hs[k] = h* + phi^k (h_last - h*), phi = al+be in (0,1); out = sqrt(hs)
    const float phi   = al + be;
    const float hstar = om / (1.0f - phi);
    const float dev   = h - hstar;
    for (int k = lane; k < P; k += 32) {
        float phik = powf(phi, (float)k);
        out[(size_t)b * P + k] = sqrtf(fmaf(phik, dev, hstar));
    }
}

extern "C" void kernel_launch(void* const* d_in, const int* in_sizes, int n_in,
                              void* d_out, int out_size, void* d_ws, size_t ws_size,
                              hipStream_t stream) {
    const float* returns = (const float*)d_in[0];
    const float* omega   = (const float*)d_in[1];
    const float* alpha   = (const float*)d_in[2];
    const float* beta    = (const float*)d_in[3];
    const float* h0p     = (const float*)d_in[4];
    const int*   plen    = (const int*)d_in[5];

    const int B = in_sizes[0] / SEQ;                 // 8192 for the reference shapes
    const int blocks = (B + WAVES_PER_BLOCK - 1) / WAVES_PER_BLOCK;

    garch_scan_forecast<<<blocks, WAVES_PER_BLOCK * 32, 0, stream>>>(
        returns, omega, alpha, beta, h0p, plen, (float*)d_out, B);
}